// MolDis_54666343743855
// MI455X (gfx1250) — compile-verified
//
#include <hip/hip_runtime.h>
#include <hip/hip_bf16.h>

// Problem size is fixed by the reference.
constexpr int N_NODES = 1048576;
constexpr int N_EDGES = 4194304;
constexpr int D = 32;
constexpr int R = 4;
constexpr int BS = 32;

typedef __attribute__((ext_vector_type(2))) float v2f;
typedef __attribute__((ext_vector_type(8))) float v8f;

// ---------------------------------------------------------------------------
// Projection kernel: per 16-node tile (one wave32):
//   hw[n, r, :]  = h[n,:] @ W[r]          (r = 0..3)
//   acc[n, :]    = h[n,:] @ W_loop + bias (self-loop term, seeds the scatter
//                                          accumulator)
// Uses V_WMMA_F32_16X16X4_F32: A = 16x4 tile of h, B = 4x16 tile of W,
// accumulate over 8 K-chunks (K = 32).
// A layout (32-bit, 16x4): lane m=lane&15, kh=lane>>4 holds
//   v0 = A[m][4kc+2kh], v1 = A[m][4kc+2kh+1]   -> one aligned 8B load.
// B layout mirrors A with M<->N. C/D layout: vgpr j -> M = j + 8*kh,
//   N = (lane&15) + 16*nt.
// ---------------------------------------------------------------------------
template <bool RELU_IN>
__global__ __launch_bounds__(256) void rgcn_proj_wmma(
    const float* __restrict__ h, const float* __restrict__ Wrel,
    const float* __restrict__ Wloop, const float* __restrict__ bias,
    float* __restrict__ hw, float* __restrict__ acc) {
  const int wave  = blockIdx.x * (blockDim.x >> 5) + (threadIdx.x >> 5);
  const int lane  = threadIdx.x & 31;
  const int m     = lane & 15;
  const int kh    = lane >> 4;
  const int node0 = wave * 16;  // grid exactly covers N_NODES/16 tiles

  // Preload all 8 A fragments for this tile's 16 rows (reused for all 5 GEMMs)
  const float* row = h + (size_t)(node0 + m) * D;
  v2f a[8];
#pragma unroll
  for (int kc = 0; kc < 8; ++kc) {
    v2f v = *reinterpret_cast<const v2f*>(row + 4 * kc + 2 * kh);
    if (RELU_IN) {
      v.x = fmaxf(v.x, 0.0f);
      v.y = fmaxf(v.y, 0.0f);
    }
    a[kc] = v;
  }

  for (int r = 0; r < R + 1; ++r) {  // r==R -> W_loop
    const float* Wm = (r < R) ? (Wrel + (size_t)r * D * D) : Wloop;
#pragma unroll
    for (int nt = 0; nt < 2; ++nt) {
      const int ncol = nt * 16 + m;
      v8f c = {};
#pragma unroll
      for (int kc = 0; kc < 8; ++kc) {
        const int k0 = 4 * kc + 2 * kh;
        v2f b;
        b.x = Wm[(size_t)k0 * D + ncol];
        b.y = Wm[(size_t)(k0 + 1) * D + ncol];
        // 8 args: (neg_a, A, neg_b, B, c_mod, C, reuse_a, reuse_b)
        c = __builtin_amdgcn_wmma_f32_16x16x4_f32(false, a[kc], false, b,
                                                  (short)0, c, false, false);
      }
      if (r < R) {
#pragma unroll
        for (int j = 0; j < 8; ++j) {
          const int M = j + 8 * kh;
          hw[((size_t)(node0 + M) * R + r) * D + ncol] = c[j];
        }
      } else {
        const float bv = bias[ncol];
#pragma unroll
        for (int j = 0; j < 8; ++j) {
          const int M = j + 8 * kh;
          acc[(size_t)(node0 + M) * D + ncol] = c[j] + bv;
        }
      }
    }
  }
}

// ---------------------------------------------------------------------------
// Edge scatter: one wave per edge (32 lanes = 32 feature channels).
// src/dst/etype loads are wave-uniform (broadcast); message row read and
// atomic adds are fully coalesced 128B accesses.
// ---------------------------------------------------------------------------
__global__ __launch_bounds__(256) void rgcn_edge_scatter(
    const float* __restrict__ hw, const int* __restrict__ src,
    const int* __restrict__ dst, const int* __restrict__ etype,
    float* __restrict__ acc) {
  const long gid = (long)blockIdx.x * blockDim.x + threadIdx.x;
  const long e = gid >> 5;
  const int d = (int)(gid & 31);
  if (e >= N_EDGES) return;
  const int s = src[e];
  const int t = dst[e];
  const int r = etype[e];
  const float v = hw[((size_t)s * R + r) * D + d];
  atomicAdd(acc + (size_t)t * D + d, v);
}

// ---------------------------------------------------------------------------
// Final fused ReLU + per-molecule mean over bs=32 atoms.
// out[mol][d] = mean_a relu(h[mol*32 + a][d])
// ---------------------------------------------------------------------------
__global__ __launch_bounds__(256) void rgcn_mean_relu(
    const float* __restrict__ h, float* __restrict__ out) {
  const int gid = blockIdx.x * blockDim.x + threadIdx.x;  // N_NODES outputs? no:
  // one thread per (molecule, channel): (N_NODES/BS) * D = 1M threads
  const int mol = gid >> 5;
  const int d = gid & 31;
  const float* p = h + (size_t)mol * BS * D + d;
  float s = 0.0f;
#pragma unroll 8
  for (int a = 0; a < BS; ++a) s += fmaxf(p[(size_t)a * D], 0.0f);
  out[(size_t)mol * D + d] = s * (1.0f / (float)BS);
}

extern "C" void kernel_launch(void* const* d_in, const int* in_sizes, int n_in,
                              void* d_out, int out_size, void* d_ws,
                              size_t ws_size, hipStream_t stream) {
  const float* x      = (const float*)d_in[0];  // [N, D]
  const float* W      = (const float*)d_in[1];  // [R, D, D]
  const float* W_loop = (const float*)d_in[2];  // [D, D]
  const float* bias   = (const float*)d_in[3];  // [D]
  const int* src      = (const int*)d_in[4];    // [E]
  const int* dst      = (const int*)d_in[5];    // [E]
  const int* etype    = (const int*)d_in[6];    // [E]
  float* out          = (float*)d_out;          // [N/BS, D]

  // Workspace layout (fully overwritten every call):
  //   hw   : N * R * D floats (512 MB)
  //   buf0 : N * D floats     (128 MB)  layer-1 pre-ReLU accumulator
  //   buf1 : N * D floats     (128 MB)  layer-2 pre-ReLU accumulator
  float* hw   = (float*)d_ws;
  float* buf0 = hw + (size_t)N_NODES * R * D;
  float* buf1 = buf0 + (size_t)N_NODES * D;

  const int proj_blocks    = (N_NODES / 16) / 8;           // 8 waves/block
  const int scatter_blocks = (int)(((long)N_EDGES * 32) / 256);
  const int mean_blocks    = (N_NODES * D / BS) / 256;

  // ----- Layer 1 -----
  rgcn_proj_wmma<false><<<proj_blocks, 256, 0, stream>>>(x, W, W_loop, bias,
                                                         hw, buf0);
  rgcn_edge_scatter<<<scatter_blocks, 256, 0, stream>>>(hw, src, dst, etype,
                                                        buf0);
  // ----- Layer 2 (ReLU fused into A-fragment load) -----
  rgcn_proj_wmma<true><<<proj_blocks, 256, 0, stream>>>(buf0, W, W_loop, bias,
                                                        hw, buf1);
  rgcn_edge_scatter<<<scatter_blocks, 256, 0, stream>>>(hw, src, dst, etype,
                                                        buf1);
  // ----- ReLU + per-molecule mean -----
  rgcn_mean_relu<<<mean_blocks, 256, 0, stream>>>(buf1, out);
}